// FiveStreamGCN_Model_75496935129423
// MI455X (gfx1250) — compile-verified
//
#include <hip/hip_runtime.h>
#include <math.h>

typedef float v2f __attribute__((ext_vector_type(2)));
typedef float v8f __attribute__((ext_vector_type(8)));

#define Bv 16
#define Tv 128
#define Jv 25
#define Fv 3
#define Hv 256
#define Ev 100000
#define NNv (Bv*Tv*Jv)   /* 51200 */
#define BTv (Bv*Tv)      /* 2048  */
#define EPSv 1e-5f

__device__ __forceinline__ float sigmoidf_(float x){ return 1.0f/(1.0f+expf(-x)); }

// ---------------------------------------------------------------------------
// fp32 GEMM on the WMMA pipe: C[M,N] = act(A[M,K] @ B + bias)
// BT==0: B is (K,N) row-major.  BT==1: B is (N,K) row-major (A @ B^T).
// One wave computes a 16x64 strip (4 accumulators) via V_WMMA_F32_16X16X4_F32,
// software-pipelined 1 deep so fragment loads overlap the matrix pipe.
// Requirements (all call sites satisfy them): M % 16 == 0, N % 16 == 0.
// K tail (K % 4 != 0, only K=3 and K=9 here) handled once, outside hot loop.
// ---------------------------------------------------------------------------
template<int BT>
__device__ __forceinline__ void gemm_load_frags(
    const float* __restrict__ Ap, const float* __restrict__ Bm,
    int kk, int N, int K, const int* nc,
    v2f& a, v2f& b0, v2f& b1, v2f& b2, v2f& b3)
{
  a.x = Ap[kk];
  a.y = Ap[kk + 1];
  if (BT == 0) {
    const float* Br0 = Bm + (long)kk * N;
    const float* Br1 = Br0 + N;
    b0.x = Br0[nc[0]]; b0.y = Br1[nc[0]];
    b1.x = Br0[nc[1]]; b1.y = Br1[nc[1]];
    b2.x = Br0[nc[2]]; b2.y = Br1[nc[2]];
    b3.x = Br0[nc[3]]; b3.y = Br1[nc[3]];
  } else {
    b0.x = Bm[(long)nc[0]*K + kk]; b0.y = Bm[(long)nc[0]*K + kk + 1];
    b1.x = Bm[(long)nc[1]*K + kk]; b1.y = Bm[(long)nc[1]*K + kk + 1];
    b2.x = Bm[(long)nc[2]*K + kk]; b2.y = Bm[(long)nc[2]*K + kk + 1];
    b3.x = Bm[(long)nc[3]*K + kk]; b3.y = Bm[(long)nc[3]*K + kk + 1];
  }
}

template<int BT>
__global__ __launch_bounds__(128) void gemm_wmma_f32_t(
    const float* __restrict__ A, int lda,
    const float* __restrict__ Bm,
    const float* __restrict__ bias,
    float* __restrict__ C, int ldc,
    int M, int N, int K, int act)
{
  int wave  = threadIdx.x >> 5;
  int lane  = threadIdx.x & 31;
  int mtile = blockIdx.x * 4 + wave;
  if (mtile * 16 >= M) return;
  int m0  = mtile << 4;
  int n0  = blockIdx.y << 6;            // 64 columns per wave
  int nh  = lane >> 4;                  // lane half -> K pair selector
  int l15 = lane & 15;

  const float* Ap = A + (long)(m0 + l15) * lda;

  int  nc[4];
  bool nv[4];
#pragma unroll
  for (int t = 0; t < 4; ++t){
    int n = n0 + (t << 4) + l15;
    nv[t] = (n0 + (t << 4)) < N;        // wave-uniform per tile (N % 16 == 0)
    nc[t] = n < N ? n : (N - 1);        // clamp: loads stay in bounds, no branches
  }

  v8f c0 = {0.f,0.f,0.f,0.f,0.f,0.f,0.f,0.f};
  v8f c1 = c0, c2 = c0, c3 = c0;

  const int K4 = K & ~3;
  const int kbase = nh << 1;            // lanes0-15: +0 ; lanes16-31: +2
  if (K4 > 0) {
    v2f a, b0, b1, b2, b3;
    gemm_load_frags<BT>(Ap, Bm, kbase, N, K, nc, a, b0, b1, b2, b3);
    for (int k0 = 4; k0 < K4; k0 += 4) {
      v2f an, bn0, bn1, bn2, bn3;       // issue next-step loads first...
      gemm_load_frags<BT>(Ap, Bm, k0 + kbase, N, K, nc, an, bn0, bn1, bn2, bn3);
      // ...then consume the previous step's fragments (loads overlap WMMA).
      c0 = __builtin_amdgcn_wmma_f32_16x16x4_f32(false, a, false, b0, (short)0, c0, false, false);
      c1 = __builtin_amdgcn_wmma_f32_16x16x4_f32(false, a, false, b1, (short)0, c1, false, false);
      c2 = __builtin_amdgcn_wmma_f32_16x16x4_f32(false, a, false, b2, (short)0, c2, false, false);
      c3 = __builtin_amdgcn_wmma_f32_16x16x4_f32(false, a, false, b3, (short)0, c3, false, false);
      a = an; b0 = bn0; b1 = bn1; b2 = bn2; b3 = bn3;
    }
    c0 = __builtin_amdgcn_wmma_f32_16x16x4_f32(false, a, false, b0, (short)0, c0, false, false);
    c1 = __builtin_amdgcn_wmma_f32_16x16x4_f32(false, a, false, b1, (short)0, c1, false, false);
    c2 = __builtin_amdgcn_wmma_f32_16x16x4_f32(false, a, false, b2, (short)0, c2, false, false);
    c3 = __builtin_amdgcn_wmma_f32_16x16x4_f32(false, a, false, b3, (short)0, c3, false, false);
  }

  if (K4 < K) {                         // one guarded tail step (K = 3 or 9)
    int kk = K4 + kbase;
    v2f a;
    a.x = (kk     < K) ? Ap[kk]     : 0.0f;
    a.y = (kk + 1 < K) ? Ap[kk + 1] : 0.0f;
    v2f bt[4];
#pragma unroll
    for (int t = 0; t < 4; ++t){
      if (BT == 0) {
        bt[t].x = (kk     < K) ? Bm[(long)(kk    )*N + nc[t]] : 0.0f;
        bt[t].y = (kk + 1 < K) ? Bm[(long)(kk + 1)*N + nc[t]] : 0.0f;
      } else {
        bt[t].x = (kk     < K) ? Bm[(long)nc[t]*K + kk    ] : 0.0f;
        bt[t].y = (kk + 1 < K) ? Bm[(long)nc[t]*K + kk + 1] : 0.0f;
      }
    }
    c0 = __builtin_amdgcn_wmma_f32_16x16x4_f32(false, a, false, bt[0], (short)0, c0, false, false);
    c1 = __builtin_amdgcn_wmma_f32_16x16x4_f32(false, a, false, bt[1], (short)0, c1, false, false);
    c2 = __builtin_amdgcn_wmma_f32_16x16x4_f32(false, a, false, bt[2], (short)0, c2, false, false);
    c3 = __builtin_amdgcn_wmma_f32_16x16x4_f32(false, a, false, bt[3], (short)0, c3, false, false);
  }

  v8f* acc[4] = { &c0, &c1, &c2, &c3 };
#pragma unroll
  for (int t = 0; t < 4; ++t){
    if (!nv[t]) continue;               // wave-uniform branch
    int ncol = nc[t];
    float bv = bias ? bias[ncol] : 0.0f;
#pragma unroll
    for (int r = 0; r < 8; ++r) {       // C layout: VGPR r -> rows r (lanes0-15), r+8 (16-31)
      long row = m0 + r + (nh << 3);
      float v = (*acc[t])[r] + bv;
      if (act == 1) v = fmaxf(v, 0.0f);
      C[row * (long)ldc + ncol] = v;
    }
  }
}

// ------------------------------- GCN pieces --------------------------------
__global__ void deg_init_k(float* deg, int n){
  int i = blockIdx.x*blockDim.x+threadIdx.x;
  if (i < n) deg[i] = 1.0f;                      // self loop
}
__global__ void deg_scatter_k(const int* __restrict__ dst, float* deg, int E, int mod){
  int e = blockIdx.x*blockDim.x+threadIdx.x;
  if (e >= E) return;
  int d = dst[e]; if (mod) d %= mod;
  atomicAdd(&deg[d], 1.0f);
}
__global__ void dinv_k(const float* deg, float* dinv, int n){
  int i = blockIdx.x*blockDim.x+threadIdx.x;
  if (i < n) dinv[i] = rsqrtf(deg[i]);
}
__global__ void gcn_scatter_k(const float* __restrict__ h, const int* __restrict__ src,
                              const int* __restrict__ dst, const float* __restrict__ dinv,
                              float* __restrict__ acc, int E, int H, int mod){
  int ch = H >> 2;
  long tot = (long)E * ch;
  for (long idx = blockIdx.x*(long)blockDim.x + threadIdx.x; idx < tot;
       idx += (long)gridDim.x*blockDim.x){
    int e = (int)(idx / ch);
    int c = ((int)(idx % ch)) << 2;
    int s = src[e], d = dst[e];
    if (mod){ s %= mod; d %= mod; }
    float w = dinv[s]*dinv[d];
    const float* hp = h + (long)s*H + c;
    float* ap = acc + (long)d*H + c;
    atomicAdd(ap+0, hp[0]*w);
    atomicAdd(ap+1, hp[1]*w);
    atomicAdd(ap+2, hp[2]*w);
    atomicAdd(ap+3, hp[3]*w);
  }
}
__global__ void gcn_finalize_k(const float* acc, const float* h, const float* dinv,
                               const float* bias, float* out, long N, int H){
  long idx = blockIdx.x*(long)blockDim.x + threadIdx.x;
  if (idx >= N*H) return;
  long node = idx / H; int c = (int)(idx % H);
  float di = dinv[node];
  out[idx] = fmaxf(acc[idx] + h[idx]*di*di + bias[c], 0.0f);
}

// ------------------------------ hyper module -------------------------------
__global__ void rowsq_k(const float* xn, float* sq){
  __shared__ float red[256];
  int row = blockIdx.x, c = threadIdx.x;
  float v = xn[(long)row*Hv + c];
  red[c] = v*v; __syncthreads();
  for (int s = 128; s > 0; s >>= 1){ if (c < s) red[c] += red[c+s]; __syncthreads(); }
  if (c == 0) sq[row] = red[0];
}
__global__ void pdist_k(const float* xn, const float* sq, float* dm){
  __shared__ float sx[Jv*Hv];
  __shared__ float ss[Jv];
  int b = blockIdx.x;
  for (int i = threadIdx.x; i < Jv*Hv; i += blockDim.x) sx[i] = xn[(long)b*Jv*Hv + i];
  if (threadIdx.x < Jv) ss[threadIdx.x] = sq[b*Jv + threadIdx.x];
  __syncthreads();
  for (int p = threadIdx.x; p < Jv*Jv; p += blockDim.x){
    int i = p / Jv, j = p % Jv;
    float dot = 0.f;
    for (int c = 0; c < Hv; ++c) dot += sx[i*Hv+c]*sx[j*Hv+c];
    float d2 = ss[i] + ss[j] - 2.f*dot;
    dm[(long)b*Jv*Jv + p] = d2 > 0.f ? sqrtf(d2) : 0.f;
  }
}
__global__ void poolfeat_k(const float* xn, float* feat){
  int idx = blockIdx.x*blockDim.x + threadIdx.x;
  if (idx >= BTv*Hv) return;
  int bt = idx / Hv, c = idx % Hv;
  float s = 0.f, m = -3.4e38f;
  for (int j = 0; j < Jv; ++j){
    float v = xn[((long)bt*Jv + j)*Hv + c];
    s += v; m = fmaxf(m, v);
  }
  feat[(long)bt*2*Hv + c]       = s / (float)Jv;
  feat[(long)bt*2*Hv + Hv + c]  = m;
}
__global__ void thr_k(const float* mid, const float* w2, const float* b2, float* thr){
  int b = blockIdx.x*blockDim.x + threadIdx.x;
  if (b >= BTv) return;
  float a = b2[0];
  for (int k = 0; k < 16; ++k) a += mid[b*16+k]*w2[k];
  thr[b] = sigmoidf_(a);
}
__global__ void edge_hg_k(const float* dm, const float* thr, const float* w1, const float* b1,
                          const float* w2, const float* b2, float* hg){
  __shared__ float W1[96], B1[32], W2[32];
  __shared__ float b2s;
  if (threadIdx.x < 96) W1[threadIdx.x] = w1[threadIdx.x];
  if (threadIdx.x < 32){ B1[threadIdx.x] = b1[threadIdx.x]; W2[threadIdx.x] = w2[threadIdx.x]; }
  if (threadIdx.x == 0) b2s = b2[0];
  __syncthreads();
  long idx = blockIdx.x*(long)blockDim.x + threadIdx.x;
  if (idx >= (long)BTv*Jv*Jv) return;
  int b = (int)(idx / (Jv*Jv));
  float d = dm[idx];
  float f0 = d, f1 = cosf(d), f2 = expf(-d);
  float a = b2s;
  for (int hh = 0; hh < 32; ++hh){
    float t = f0*W1[hh] + f1*W1[32+hh] + f2*W1[64+hh] + B1[hh];
    t = fmaxf(t, 0.f);
    a += t*W2[hh];
  }
  float ew = sigmoidf_(a);
  hg[idx] = (d < thr[b]) ? ew : 0.f;
}
__global__ void msgmean_T_k(const float* hg, const float* y, float* Ee){
  __shared__ float hl[Jv*Jv];
  __shared__ float cs[Jv];
  int b = blockIdx.x;
  for (int i = threadIdx.x; i < Jv*Jv; i += blockDim.x) hl[i] = hg[(long)b*Jv*Jv + i];
  __syncthreads();
  if (threadIdx.x < Jv){
    float t = 0.f;
    for (int j = 0; j < Jv; ++j) t += hl[j*Jv + threadIdx.x];
    cs[threadIdx.x] = t;
  }
  __syncthreads();
  for (int o = threadIdx.x; o < Jv*Hv; o += blockDim.x){
    int i = o / Hv, c = o % Hv;
    float a = 0.f;
    for (int j = 0; j < Jv; ++j) a += hl[j*Jv + i]*y[((long)b*Jv + j)*Hv + c];
    float s = cs[i];
    Ee[((long)b*Jv + i)*Hv + c] = (s != 0.f) ? a/s : 0.f;
  }
}
__global__ void msgmean_k(const float* hg, const float* Ee, const float* xn, float* out){
  __shared__ float hl[Jv*Jv];
  __shared__ float rs[Jv];
  int b = blockIdx.x;
  for (int i = threadIdx.x; i < Jv*Jv; i += blockDim.x) hl[i] = hg[(long)b*Jv*Jv + i];
  __syncthreads();
  if (threadIdx.x < Jv){
    float t = 0.f;
    for (int j = 0; j < Jv; ++j) t += hl[threadIdx.x*Jv + j];
    rs[threadIdx.x] = t;
  }
  __syncthreads();
  for (int o = threadIdx.x; o < Jv*Hv; o += blockDim.x){
    int i = o / Hv, c = o % Hv;
    float a = 0.f;
    for (int j = 0; j < Jv; ++j) a += hl[i*Jv + j]*Ee[((long)b*Jv + j)*Hv + c];
    float s = rs[i];
    long oi = ((long)b*Jv + i)*Hv + c;
    out[oi] = ((s != 0.f) ? a/s : 0.f) + xn[oi];
  }
}
__global__ void bn_stats_k(const float* x, float* stats, long rows, int rowsPerBlock){
  int c = threadIdx.x;
  long r0 = (long)blockIdx.x * rowsPerBlock;
  long r1 = r0 + rowsPerBlock; if (r1 > rows) r1 = rows;
  float s = 0.f, ss = 0.f;
  for (long r = r0; r < r1; ++r){ float v = x[r*Hv + c]; s += v; ss += v*v; }
  atomicAdd(&stats[c],      s);
  atomicAdd(&stats[Hv + c], ss);
}
__global__ void bn_apply_k(float* xio, const float* stats, const float* g, const float* bb, long rows){
  long idx = blockIdx.x*(long)blockDim.x + threadIdx.x;
  if (idx >= rows*Hv) return;
  int c = (int)(idx % Hv);
  float mu  = stats[c] / (float)rows;
  float var = stats[Hv + c] / (float)rows - mu*mu;
  float v = xio[idx];
  float o = g[c]*(v - mu)*rsqrtf(var + EPSv) + bb[c];
  xio[idx] = o * sigmoidf_(o);           // silu
}

// ------------------------------ transformer --------------------------------
__global__ void attn_k(const float* qkv, float* o){
  __shared__ float qs[Jv*64], ks[Jv*64], vs[Jv*64], sc[Jv*Jv];
  int b = blockIdx.x, h = blockIdx.y;
  for (int idx = threadIdx.x; idx < Jv*64; idx += blockDim.x){
    int s = idx / 64, d = idx % 64;
    long row = ((long)b*Jv + s)*768;
    qs[idx] = qkv[row +       h*64 + d];
    ks[idx] = qkv[row + 256 + h*64 + d];
    vs[idx] = qkv[row + 512 + h*64 + d];
  }
  __syncthreads();
  for (int p = threadIdx.x; p < Jv*Jv; p += blockDim.x){
    int i = p / Jv, j = p % Jv;
    float dot = 0.f;
    for (int d = 0; d < 64; ++d) dot += qs[i*64+d]*ks[j*64+d];
    sc[p] = dot * 0.125f;                // 1/sqrt(64)
  }
  __syncthreads();
  if (threadIdx.x < Jv){
    int i = threadIdx.x;
    float mx = -3.4e38f;
    for (int j = 0; j < Jv; ++j) mx = fmaxf(mx, sc[i*Jv+j]);
    float sum = 0.f;
    for (int j = 0; j < Jv; ++j){ float e = expf(sc[i*Jv+j]-mx); sc[i*Jv+j] = e; sum += e; }
    float inv = 1.f/sum;
    for (int j = 0; j < Jv; ++j) sc[i*Jv+j] *= inv;
  }
  __syncthreads();
  for (int idx = threadIdx.x; idx < Jv*64; idx += blockDim.x){
    int s = idx / 64, d = idx % 64;
    float a = 0.f;
    for (int j = 0; j < Jv; ++j) a += sc[s*Jv+j]*vs[j*64+d];
    o[((long)b*Jv + s)*Hv + h*64 + d] = a;
  }
}
__global__ void add_ln_k(const float* x, const float* y, const float* g, const float* bb,
                         float* out, int H, int act){
  extern __shared__ float red[];
  int row = blockIdx.x, c = threadIdx.x;
  long base = (long)row*H;
  float v = x[base+c] + (y ? y[base+c] : 0.f);
  red[c] = v; __syncthreads();
  for (int s = H>>1; s > 0; s >>= 1){ if (c < s) red[c] += red[c+s]; __syncthreads(); }
  float mu = red[0] / (float)H;
  __syncthreads();
  float dv = v - mu;
  red[c] = dv*dv; __syncthreads();
  for (int s = H>>1; s > 0; s >>= 1){ if (c < s) red[c] += red[c+s]; __syncthreads(); }
  float var = red[0] / (float)H;
  float o = g[c]*dv*rsqrtf(var + EPSv) + bb[c];
  if (act) o = fmaxf(o, 0.f);
  out[base+c] = o;
}

// --------------------------------- topo ------------------------------------
__global__ void topo_k(const float* x, float* acc){
  int idx = blockIdx.x*blockDim.x + threadIdx.x;
  if (idx >= NNv) return;
  int j  = idx % Jv;
  int bt = idx / Jv;
  int b  = bt / Tv;
  const float* base = x + (long)bt*Jv*3;
  float px = base[j*3+0], py = base[j*3+1], pz = base[j*3+2];
  float best[5] = {3.4e38f,3.4e38f,3.4e38f,3.4e38f,3.4e38f};
  for (int jj = 0; jj < Jv; ++jj){
    float dx = px-base[jj*3], dy = py-base[jj*3+1], dz = pz-base[jj*3+2];
    float d2 = dx*dx + dy*dy + dz*dz;
    float d = d2 > 0.f ? sqrtf(d2) : 0.f;
    if (d < best[4]){
      best[4] = d;
      for (int q = 4; q > 0; --q){
        if (best[q] < best[q-1]){ float t = best[q-1]; best[q-1] = best[q]; best[q] = t; }
      }
    }
  }
  float sd[4] = {best[1], best[2], best[3], best[4]};
  const float invT = 1.0f / (float)Tv;
  float* ap = acc + ((long)b*Jv + j)*9;
  for (int k = 2; k <= 4; ++k){
    float s = 0.f; for (int i = 0; i < k; ++i) s += sd[i];
    float mean = s / (float)k;
    float ss = 0.f; for (int i = 0; i < k; ++i){ float dd = sd[i]-mean; ss += dd*dd; }
    float stdv = sqrtf(ss / (float)(k-1));
    atomicAdd(ap + (k-2)*3 + 0, mean*invT);
    atomicAdd(ap + (k-2)*3 + 1, stdv*invT);
    atomicAdd(ap + (k-2)*3 + 2, sd[0]*invT);
  }
}
__global__ void tfmean_k(const float* t2, float* tf){
  int idx = blockIdx.x*blockDim.x + threadIdx.x;
  if (idx >= Bv*Hv) return;
  int b = idx / Hv, c = idx % Hv;
  float s = 0.f;
  for (int j = 0; j < Jv; ++j) s += t2[((long)b*Jv + j)*Hv + c];
  tf[idx] = s / (float)Jv;
}

// ------------------------------ fusion / GRU -------------------------------
__global__ void comb_k(const float* sk, const float* tf, float* comb){
  const int CW = Jv*Hv + Hv;                    // 6656
  long idx = blockIdx.x*(long)blockDim.x + threadIdx.x;
  if (idx >= (long)BTv*CW) return;
  int bt = (int)(idx / CW), c = (int)(idx % CW);
  float v;
  if (c < Jv*Hv) v = sk[((long)bt*Jv + c/Hv)*Hv + (c%Hv)];
  else           v = tf[(bt/Tv)*Hv + (c - Jv*Hv)];
  comb[idx] = v;
}
__global__ void gru_gate_k(const float* gi, const float* gh, float* h, float* ys, int T){
  int idx = blockIdx.x*blockDim.x + threadIdx.x;
  if (idx >= Bv*Hv) return;
  int b = idx / Hv, c = idx % Hv;
  const float* gib = gi + (long)b*3*Hv;
  const float* ghb = gh + (long)b*3*Hv;
  float r = sigmoidf_(gib[c]        + ghb[c]);
  float z = sigmoidf_(gib[Hv+c]     + ghb[Hv+c]);
  float n = tanhf   (gib[2*Hv+c] + r*ghb[2*Hv+c]);
  float hn = (1.0f - z)*n + z*h[idx];
  h[idx] = hn;
  ys[(long)b*T*Hv + c] = hn;
}
__global__ void fc_k(const float* h, const float* w, const float* b, float* out, int O){
  int idx = blockIdx.x*blockDim.x + threadIdx.x;
  if (idx >= Bv*O) return;
  int bb = idx / O, o = idx % O;
  float acc = b[o];
  for (int k = 0; k < Hv; ++k) acc += h[bb*Hv + k]*w[k*O + o];
  out[idx] = acc;
}

// ---------------------------------------------------------------------------
extern "C" void kernel_launch(void* const* d_in, const int* in_sizes, int n_in,
                              void* d_out, int out_size, void* d_ws, size_t ws_size,
                              hipStream_t stream)
{
  (void)in_sizes; (void)n_in; (void)out_size;
  const float* x  = (const float*)d_in[0];
  const int*   ei = (const int*)d_in[1];
  const int* src = ei;
  const int* dst = ei + Ev;
  const float *gcn1_w=(const float*)d_in[2],  *gcn1_b=(const float*)d_in[3];
  const float *gcn2_w=(const float*)d_in[4],  *gcn2_b=(const float*)d_in[5];
  const float *hyp_w =(const float*)d_in[6],  *hyp_b =(const float*)d_in[7];
  const float *thr_w1=(const float*)d_in[8],  *thr_b1=(const float*)d_in[9];
  const float *thr_w2=(const float*)d_in[10], *thr_b2=(const float*)d_in[11];
  const float *edg_w1=(const float*)d_in[12], *edg_b1=(const float*)d_in[13];
  const float *edg_w2=(const float*)d_in[14], *edg_b2=(const float*)d_in[15];
  const float *bn_g  =(const float*)d_in[16], *bn_b  =(const float*)d_in[17];
  const float *ain_w =(const float*)d_in[18], *ain_b =(const float*)d_in[19];
  const float *aout_w=(const float*)d_in[20], *aout_b=(const float*)d_in[21];
  const float *ln1_g =(const float*)d_in[22], *ln1_b =(const float*)d_in[23];
  const float *ln2_g =(const float*)d_in[24], *ln2_b =(const float*)d_in[25];
  const float *ff_w1 =(const float*)d_in[26], *ff_b1 =(const float*)d_in[27];
  const float *ff_w2 =(const float*)d_in[28], *ff_b2 =(const float*)d_in[29];
  const float *tg1_w =(const float*)d_in[30], *tg1_b =(const float*)d_in[31];
  const float *tg2_w =(const float*)d_in[32], *tg2_b =(const float*)d_in[33];
  const float *fus_w1=(const float*)d_in[34], *fus_b1=(const float*)d_in[35];
  const float *fln_g =(const float*)d_in[36], *fln_b =(const float*)d_in[37];
  const float *fus_w2=(const float*)d_in[38], *fus_b2=(const float*)d_in[39];
  const float *g0_wih=(const float*)d_in[40], *g0_whh=(const float*)d_in[41];
  const float *g0_bih=(const float*)d_in[42], *g0_bhh=(const float*)d_in[43];
  const float *g1_wih=(const float*)d_in[44], *g1_whh=(const float*)d_in[45];
  const float *g1_bih=(const float*)d_in[46], *g1_bhh=(const float*)d_in[47];
  const float *fc_w  =(const float*)d_in[48], *fc_b  =(const float*)d_in[49];

  // ---- workspace arena ----
  float* base = (float*)d_ws;
  size_t off = 0;
  auto alloc = [&](size_t n)->float*{ float* p = base + off; off += (n + 255) & ~(size_t)255; return p; };
  float* deg   = alloc(NNv);
  float* dinv  = alloc(NNv);
  float* deg2  = alloc(400);
  float* dinv2 = alloc(400);
  float* S1 = alloc((size_t)NNv*Hv);
  float* S2 = alloc((size_t)NNv*Hv);
  float* S3 = alloc((size_t)NNv*Hv);
  float* S4 = alloc((size_t)NNv*Hv);
  float* SA = alloc((size_t)NNv*1024);
  float* sq   = alloc((size_t)BTv*Jv);
  float* dm   = alloc((size_t)BTv*Jv*Jv);
  float* hg   = alloc((size_t)BTv*Jv*Jv);
  float* feat = alloc((size_t)BTv*2*Hv);
  float* thrm = alloc((size_t)BTv*16);
  float* thr  = alloc(BTv);
  float* stats= alloc(2*Hv);
  float* topo = alloc(400*9);
  float* tA   = alloc(400*Hv);
  float* tB   = alloc(400*Hv);
  float* tf   = alloc(Bv*Hv);
  float* comb = alloc((size_t)BTv*(Jv*Hv+Hv));
  float* fus1 = alloc((size_t)BTv*2*Hv);
  float* fus  = alloc((size_t)BTv*Hv);
  float* gh0  = alloc(Bv*Hv);
  float* gh1  = alloc(Bv*Hv);
  float* gi   = alloc(Bv*3*Hv);
  float* gg   = alloc(Bv*3*Hv);
  float* gy0  = alloc((size_t)BTv*Hv);
  float* gy1  = alloc((size_t)BTv*Hv);
  if (off*sizeof(float) > ws_size) return;

  auto gemm = [&](const float* A, int lda, const float* Bm, int bT, const float* bias,
                  float* C, int ldc, int M, int N, int K, int act){
    dim3 g((unsigned)((M/16 + 3)/4), (unsigned)((N + 63)/64));
    if (bT) gemm_wmma_f32_t<1><<<g, 128, 0, stream>>>(A, lda, Bm, bias, C, ldc, M, N, K, act);
    else    gemm_wmma_f32_t<0><<<g, 128, 0, stream>>>(A, lda, Bm, bias, C, ldc, M, N, K, act);
  };
  auto grid1 = [](long n){ return (unsigned)((n + 255)/256); };

  // ================= GCN stream =================
  deg_init_k   <<<grid1(NNv), 256, 0, stream>>>(deg, NNv);
  deg_scatter_k<<<grid1(Ev),  256, 0, stream>>>(dst, deg, Ev, 0);
  dinv_k       <<<grid1(NNv), 256, 0, stream>>>(deg, dinv, NNv);

  gemm(x, Fv, gcn1_w, 0, nullptr, S1, Hv, NNv, Hv, Fv, 0);                  // h = x @ W1
  hipMemsetAsync(S2, 0, (size_t)NNv*Hv*sizeof(float), stream);
  gcn_scatter_k<<<grid1((long)Ev*(Hv/4)), 256, 0, stream>>>(S1, src, dst, dinv, S2, Ev, Hv, 0);
  gcn_finalize_k<<<grid1((long)NNv*Hv), 256, 0, stream>>>(S2, S1, dinv, gcn1_b, S2, NNv, Hv);

  gemm(S2, Hv, gcn2_w, 0, nullptr, S1, Hv, NNv, Hv, Hv, 0);                 // h = g1 @ W2
  hipMemsetAsync(S3, 0, (size_t)NNv*Hv*sizeof(float), stream);
  gcn_scatter_k<<<grid1((long)Ev*(Hv/4)), 256, 0, stream>>>(S1, src, dst, dinv, S3, Ev, Hv, 0);
  gcn_finalize_k<<<grid1((long)NNv*Hv), 256, 0, stream>>>(S3, S1, dinv, gcn2_b, S3, NNv, Hv);  // xn = S3

  // ================= hyper module =================
  rowsq_k   <<<NNv, 256, 0, stream>>>(S3, sq);
  pdist_k   <<<BTv, 256, 0, stream>>>(S3, sq, dm);
  poolfeat_k<<<grid1((long)BTv*Hv), 256, 0, stream>>>(S3, feat);
  gemm(feat, 2*Hv, thr_w1, 0, thr_b1, thrm, 16, BTv, 16, 2*Hv, 1);
  thr_k     <<<grid1(BTv), 256, 0, stream>>>(thrm, thr_w2, thr_b2, thr);
  edge_hg_k <<<grid1((long)BTv*Jv*Jv), 256, 0, stream>>>(dm, thr, edg_w1, edg_b1, edg_w2, edg_b2, hg);
  gemm(S3, Hv, hyp_w, 0, hyp_b, S1, Hv, NNv, Hv, Hv, 0);                    // y
  msgmean_T_k<<<BTv, 256, 0, stream>>>(hg, S1, S2);                         // Ee
  msgmean_k  <<<BTv, 256, 0, stream>>>(hg, S2, S3, S4);                     // out(+xn)
  hipMemsetAsync(stats, 0, 2*Hv*sizeof(float), stream);
  bn_stats_k <<<200, 256, 0, stream>>>(S4, stats, NNv, 256);
  bn_apply_k <<<grid1((long)NNv*Hv), 256, 0, stream>>>(S4, stats, bn_g, bn_b, NNv);  // xb = S4

  // ================= transformer =================
  gemm(S4, Hv, ain_w, 1, ain_b, SA, 768, NNv, 768, Hv, 0);                  // qkv
  attn_k<<<dim3(BTv, 4), 256, 0, stream>>>(SA, S1);                          // attention -> S1
  gemm(S1, Hv, aout_w, 1, aout_b, S2, Hv, NNv, Hv, Hv, 0);                  // out proj
  add_ln_k<<<NNv, Hv, Hv*sizeof(float), stream>>>(S4, S2, ln1_g, ln1_b, S1, Hv, 0);  // h
  gemm(S1, Hv, ff_w1, 0, ff_b1, SA, 1024, NNv, 1024, Hv, 1);                // ff1 + relu
  gemm(SA, 1024, ff_w2, 0, ff_b2, S2, Hv, NNv, Hv, 1024, 0);                // ff2
  add_ln_k<<<NNv, Hv, Hv*sizeof(float), stream>>>(S1, S2, ln2_g, ln2_b, S3, Hv, 0);  // sk = S3

  // ================= topo stream =================
  hipMemsetAsync(topo, 0, 400*9*sizeof(float), stream);
  topo_k<<<grid1(NNv), 256, 0, stream>>>(x, topo);
  deg_init_k   <<<2, 256, 0, stream>>>(deg2, 400);
  deg_scatter_k<<<grid1(Ev), 256, 0, stream>>>(dst, deg2, Ev, Jv);
  dinv_k       <<<2, 256, 0, stream>>>(deg2, dinv2, 400);

  gemm(topo, 9, tg1_w, 0, nullptr, tA, Hv, 400, Hv, 9, 0);
  hipMemsetAsync(tB, 0, 400*Hv*sizeof(float), stream);
  gcn_scatter_k<<<grid1((long)Ev*(Hv/4)), 256, 0, stream>>>(tA, src, dst, dinv2, tB, Ev, Hv, Jv);
  gcn_finalize_k<<<grid1((long)400*Hv), 256, 0, stream>>>(tB, tA, dinv2, tg1_b, tB, 400, Hv);

  gemm(tB, Hv, tg2_w, 0, nullptr, tA, Hv, 400, Hv, Hv, 0);
  hipMemsetAsync(tB, 0, 400*Hv*sizeof(float), stream);
  gcn_scatter_k<<<grid1((long)Ev*(Hv/4)), 256, 0, stream>>>(tA, src, dst, dinv2, tB, Ev, Hv, Jv);
  gcn_finalize_k<<<grid1((long)400*Hv), 256, 0, stream>>>(tB, tA, dinv2, tg2_b, tB, 400, Hv);
  tfmean_k<<<grid1(Bv*Hv), 256, 0, stream>>>(tB, tf);

  // ================= fusion =================
  comb_k<<<grid1((long)BTv*(Jv*Hv+Hv)), 256, 0, stream>>>(S3, tf, comb);
  gemm(comb, Jv*Hv+Hv, fus_w1, 0, fus_b1, fus1, 2*Hv, BTv, 2*Hv, Jv*Hv+Hv, 0);
  add_ln_k<<<BTv, 2*Hv, 2*Hv*sizeof(float), stream>>>(fus1, nullptr, fln_g, fln_b, fus1, 2*Hv, 1);
  gemm(fus1, 2*Hv, fus_w2, 0, fus_b2, fus, Hv, BTv, Hv, 2*Hv, 0);

  // ================= GRU x2 =================
  hipMemsetAsync(gh0, 0, Bv*Hv*sizeof(float), stream);
  for (int t = 0; t < Tv; ++t){
    gemm(fus + (size_t)t*Hv, Tv*Hv, g0_wih, 1, g0_bih, gi, 3*Hv, Bv, 3*Hv, Hv, 0);
    gemm(gh0, Hv,            g0_whh, 1, g0_bhh, gg, 3*Hv, Bv, 3*Hv, Hv, 0);
    gru_gate_k<<<grid1(Bv*Hv), 256, 0, stream>>>(gi, gg, gh0, gy0 + (size_t)t*Hv, Tv);
  }
  hipMemsetAsync(gh1, 0, Bv*Hv*sizeof(float), stream);
  for (int t = 0; t < Tv; ++t){
    gemm(gy0 + (size_t)t*Hv, Tv*Hv, g1_wih, 1, g1_bih, gi, 3*Hv, Bv, 3*Hv, Hv, 0);
    gemm(gh1, Hv,            g1_whh, 1, g1_bhh, gg, 3*Hv, Bv, 3*Hv, Hv, 0);
    gru_gate_k<<<grid1(Bv*Hv), 256, 0, stream>>>(gi, gg, gh1, gy1 + (size_t)t*Hv, Tv);
  }

  // ================= head =================
  fc_k<<<1, 256, 0, stream>>>(gh1, fc_w, fc_b, (float*)d_out, 10);
}